// Contamination_85572928405916
// MI455X (gfx1250) — compile-verified
//
#include <hip/hip_runtime.h>

// Problem constants from the reference setup: B=8, C=32, H=W=512, k=8.
constexpr int KK    = 8;     // dilation / block size
constexpr int HH    = 512;
constexpr int WW    = 512;
constexpr int W4    = WW / 4;        // 128 float4 per row
constexpr int STRIP = 16;            // output rows per workgroup
constexpr int LROWS = STRIP + 2*KK;  // 32 staged rows (strip + vertical halo)
constexpr int NTHR  = 256;           // 8 waves (wave32)

typedef float f4  __attribute__((ext_vector_type(4)));
typedef int   v4i __attribute__((ext_vector_type(4)));

// ---- CDNA5 async global->LDS copy of 16 bytes per lane -----------------
__device__ __forceinline__ void async_copy_b128(const float* gsrc, float* ldst) {
#if __has_builtin(__builtin_amdgcn_global_load_async_to_lds_b128)
  typedef __attribute__((address_space(1))) v4i* gp_t;  // global (device) int4*
  typedef __attribute__((address_space(3))) v4i* lp_t;  // LDS (shared) int4*
  __builtin_amdgcn_global_load_async_to_lds_b128(
      (gp_t)(size_t)gsrc, (lp_t)(size_t)ldst, 0, 0);
#else
  unsigned           lo = (unsigned)(size_t)ldst;       // low 32 bits = LDS offset
  unsigned long long ga = (unsigned long long)(size_t)gsrc;
  asm volatile("global_load_async_to_lds_b128 %0, %1, off"
               :: "v"(lo), "v"(ga) : "memory");
#endif
}

__device__ __forceinline__ void wait_async0() {
#if __has_builtin(__builtin_amdgcn_s_wait_asynccnt)
  __builtin_amdgcn_s_wait_asynccnt(0);
#else
  asm volatile("s_wait_asynccnt 0" ::: "memory");
#endif
}

__device__ __forceinline__ f4 lds_f4(const float* t, int lrow, int c4) {
  return *(const f4*)&t[lrow * WW + c4 * 4];
}

__global__ __launch_bounds__(NTHR)
void contamination_kernel(const float* __restrict__ x, float* __restrict__ out) {
  __shared__ __align__(16) float tile[LROWS * WW];   // 64 KB

  const int    tid   = threadIdx.x;
  const int    strip = blockIdx.x;                   // 0..31 (adjacent strips adjacent in L2)
  const size_t plane = blockIdx.y;                   // 0..B*C-1
  const size_t base  = plane * (size_t)(HH * WW);
  const int    h0    = strip * STRIP;                // first output row of this strip

  // ---- Stage LROWS x WW floats (global rows h0-8 .. h0+23) via async DMA ----
  // 32 rows * 128 float4 = 4096 loads; 16 per thread; guard is wave-uniform.
#pragma unroll
  for (int j = 0; j < (LROWS * W4) / NTHR; ++j) {
    const int idx  = tid + j * NTHR;       // float4 index into LDS tile
    const int lrow = idx / W4;             // 0..31
    const int c4   = idx % W4;
    const int grow = h0 - KK + lrow;       // global row (may be out of range)
    if (grow >= 0 && grow < HH) {
      async_copy_b128(x + base + (size_t)grow * WW + c4 * 4, &tile[idx * 4]);
    }
  }
  wait_async0();
  __syncthreads();

  // ---- Compute: dilated 3x3 stencil served entirely from LDS ----
  const int c4   = tid % W4;               // float4 column 0..127
  const int rofs = tid / W4;               // 0 or 1

  const bool lf = (c4 >= 2);               // w-8 in bounds for all 4 lanes of the f4
  const bool rt = (c4 <= W4 - 3);          // w+8 in bounds
  const int  cL = lf ? c4 - 2 : c4;
  const int  cR = rt ? c4 + 2 : c4;
  const float nh = 1.0f + (lf ? 1.0f : 0.0f) + (rt ? 1.0f : 0.0f);

#pragma unroll
  for (int j = 0; j < STRIP / 2; ++j) {
    const int  r  = rofs + 2 * j;          // output row within strip
    const int  h  = h0 + r;                // global row
    const int  lr = r + KK;                // LDS row of the center
    const bool up = (h - KK) >= 0;
    const bool dn = (h + KK) < HH;
    const float nv  = 1.0f + (up ? 1.0f : 0.0f) + (dn ? 1.0f : 0.0f);
    const float inv = 0.2f / (nv * nh - 1.0f);

    // vertical 3-tap sums for the three column groups
    f4 center = lds_f4(tile, lr, c4);
    f4 vC = center;
    if (up) vC += lds_f4(tile, lr - KK, c4);
    if (dn) vC += lds_f4(tile, lr + KK, c4);

    f4 sAll = vC;
    if (lf) {
      f4 vL = lds_f4(tile, lr, cL);
      if (up) vL += lds_f4(tile, lr - KK, cL);
      if (dn) vL += lds_f4(tile, lr + KK, cL);
      sAll += vL;
    }
    if (rt) {
      f4 vR = lds_f4(tile, lr, cR);
      if (up) vR += lds_f4(tile, lr - KK, cR);
      if (dn) vR += lds_f4(tile, lr + KK, cR);
      sAll += vR;
    }

    const f4 neigh = sAll - center;
    const f4 o     = center * 0.8f + neigh * inv;
    *(f4*)&out[base + (size_t)h * WW + c4 * 4] = o;
  }
}

extern "C" void kernel_launch(void* const* d_in, const int* in_sizes, int n_in,
                              void* d_out, int out_size, void* d_ws, size_t ws_size,
                              hipStream_t stream) {
  const float* x   = (const float*)d_in[0];
  float*       out = (float*)d_out;
  const int planes = in_sizes[0] / (HH * WW);        // B*C = 256
  dim3 grid(HH / STRIP, planes);                     // (32, 256) workgroups
  contamination_kernel<<<grid, NTHR, 0, stream>>>(x, out);
}